// SVmodel_18554258718860
// MI455X (gfx1250) — compile-verified
//
#include <hip/hip_runtime.h>
#include <hip/hip_bf16.h>

#define NN 50000      // nodes
#define NE 800000     // edges

typedef __attribute__((ext_vector_type(2))) float v2f;
typedef __attribute__((ext_vector_type(8))) float v8f;

// ---------------------------------------------------------------------------
// Degree / normalization kernels
// ---------------------------------------------------------------------------
__global__ void fill1_kernel(float* __restrict__ p, int n) {
  int i = blockIdx.x * blockDim.x + threadIdx.x;
  if (i < n) p[i] = 1.0f;   // deg of A+I starts at 1 (self loop)
}

__global__ void accum_deg_kernel(const int* __restrict__ dst, float* deg, int e) {
  int i = blockIdx.x * blockDim.x + threadIdx.x;
  if (i < e) atomicAdd(&deg[dst[i]], 1.0f);
}

__global__ void rsqrt_kernel(float* __restrict__ p, int n) {
  int i = blockIdx.x * blockDim.x + threadIdx.x;
  if (i < n) p[i] = rsqrtf(p[i]);
}

// ---------------------------------------------------------------------------
// WMMA f32 GEMM: C[M,N] = A[M,K] @ B[K,N] with fused epilogues.
//   EPI 1: C = acc + bias                        (final linear)
//   EPI 2: C = elu(acc + bias)                   (projection hidden)
//   EPI 3: v = acc * rowscale[row]; C=v; C2=v    (conv: y = xW * dinv, dual store)
// One 16x16 tile per wave32; 8 waves/block cover 128 columns.
// ---------------------------------------------------------------------------
template<int EPI>
__global__ __launch_bounds__(256) void gemm_wmma_f32(
    const float* __restrict__ A, const float* __restrict__ B,
    const float* __restrict__ bias, const float* __restrict__ rowscale,
    float* __restrict__ C, float* __restrict__ C2,
    int M, int N, int K)
{
  const int wave = threadIdx.x >> 5;
  const int lane = threadIdx.x & 31;
  const int l16  = lane & 15;
  const int hi   = lane >> 4;                 // 0: K{0,1}, 1: K{2,3}
  const int m0   = blockIdx.x << 4;
  const int n0   = (blockIdx.y << 7) + (wave << 4);

  // A fragment base: row (m0+l16), K offset 2*hi  (ISA 32-bit A 16x4 layout)
  const float* Ap = A + (size_t)(m0 + l16) * K + (hi << 1);
  // B fragment base: row K offset 2*hi, col (n0+l16)
  const float* Bp = B + (size_t)(hi << 1) * N + (n0 + l16);

  v8f acc = {};
  #pragma unroll 4
  for (int k = 0; k < K; k += 4) {
    v2f a = *(const v2f*)(Ap + k);            // A[m][k+2hi], A[m][k+2hi+1]
    v2f b;
    b.x = Bp[(size_t)(k    ) * N];            // B[k+2hi  ][n]
    b.y = Bp[(size_t)(k + 1) * N];            // B[k+2hi+1][n]
    acc = __builtin_amdgcn_wmma_f32_16x16x4_f32(
        /*neg_a=*/false, a, /*neg_b=*/false, b,
        /*c_mod=*/(short)0, acc, /*reuse_a=*/false, /*reuse_b=*/false);
  }

  // C/D layout: lanes 0-15 -> rows m0..m0+7, lanes 16-31 -> rows m0+8..m0+15
  const int col = n0 + l16;
  const int r0  = m0 + (hi << 3);
  float bv = 0.0f;
  if (EPI == 1 || EPI == 2) bv = bias[col];
  #pragma unroll
  for (int j = 0; j < 8; ++j) {
    float v = acc[j];
    size_t off = (size_t)(r0 + j) * N + col;
    if (EPI == 3) {
      v *= rowscale[r0 + j];
      C[off]  = v;      // message buffer y
      C2[off] = v;      // accumulator init (= self-loop term)
    } else {
      v += bv;
      if (EPI == 2) v = v > 0.0f ? v : expm1f(v);   // ELU(alpha=1)
      C[off] = v;
    }
  }
}

// ---------------------------------------------------------------------------
// Edge scatter: acc[dst] += y[src]  (dinv factors pre-folded into y / finish)
// One thread per (edge, 4-float chunk); consecutive threads -> consecutive
// features of the same edge for coalesced float4 gathers + grouped atomics.
// ---------------------------------------------------------------------------
__global__ void scatter_kernel(const float* __restrict__ y,
                               const int* __restrict__ src,
                               const int* __restrict__ dst,
                               float* acc, int C4, int total) {
  int i = blockIdx.x * blockDim.x + threadIdx.x;
  if (i >= total) return;
  int e = i / C4, q = i - e * C4;
  int s = src[e], d = dst[e];
  const int C = C4 << 2;
  float4 v = *((const float4*)(y + (size_t)s * C) + q);
  float* o = acc + (size_t)d * C + (q << 2);
  atomicAdd(o + 0, v.x);
  atomicAdd(o + 1, v.y);
  atomicAdd(o + 2, v.z);
  atomicAdd(o + 3, v.w);
}

// buf = relu(dinv[row] * buf + bias[col])
__global__ void finish_kernel(float* __restrict__ buf,
                              const float* __restrict__ dinv,
                              const float* __restrict__ bias,
                              int C, int total) {
  int i = blockIdx.x * blockDim.x + threadIdx.x;
  if (i >= total) return;
  int node = i / C;
  int f = i - node * C;
  buf[i] = fmaxf(fmaf(dinv[node], buf[i], bias[f]), 0.0f);
}

// ---------------------------------------------------------------------------
extern "C" void kernel_launch(void* const* d_in, const int* in_sizes, int n_in,
                              void* d_out, int out_size, void* d_ws, size_t ws_size,
                              hipStream_t stream) {
  const float* x    = (const float*)d_in[0];
  const int*   ei   = (const int*)d_in[1];
  const float* W1   = (const float*)d_in[2];
  const float* b1   = (const float*)d_in[3];
  const float* W2   = (const float*)d_in[4];
  const float* b2   = (const float*)d_in[5];
  const float* fcW1 = (const float*)d_in[6];
  const float* fcb1 = (const float*)d_in[7];
  const float* fcW2 = (const float*)d_in[8];
  const float* fcb2 = (const float*)d_in[9];
  const int* src = ei;            // edge_index[0]
  const int* dst = ei + NE;       // edge_index[1]

  float* ws   = (float*)d_ws;
  float* dinv = ws;                         // 50000 floats (padded to 50048)
  float* bufA = ws + 50048;                 // 50000*256
  float* bufB = bufA + (size_t)NN * 256;    // 50000*256
  float* y2   = bufA;                       // conv2 messages [50000,128]
  float* zacc = bufA + (size_t)NN * 128;    // conv2 accumulator [50000,128]
  float* outp = (float*)d_out;

  // --- dinv = rsqrt(deg(A+I)) ---
  fill1_kernel    <<<(NN + 255) / 256, 256, 0, stream>>>(dinv, NN);
  accum_deg_kernel<<<(NE + 255) / 256, 256, 0, stream>>>(dst, dinv, NE);
  rsqrt_kernel    <<<(NN + 255) / 256, 256, 0, stream>>>(dinv, NN);

  // --- conv1: y1 = (x@W1)*dinv -> bufA, acc init -> bufB ---
  gemm_wmma_f32<3><<<dim3(NN / 16, 2), 256, 0, stream>>>(
      x, W1, nullptr, dinv, bufA, bufB, NN, 256, 128);
  scatter_kernel<<<(NE * 64 + 255) / 256, 256, 0, stream>>>(
      bufA, src, dst, bufB, 64, NE * 64);
  finish_kernel<<<(NN * 256 + 255) / 256, 256, 0, stream>>>(
      bufB, dinv, b1, 256, NN * 256);                        // h in bufB

  // --- conv2: y2 = (h@W2)*dinv -> halves of bufA ---
  gemm_wmma_f32<3><<<dim3(NN / 16, 1), 256, 0, stream>>>(
      bufB, W2, nullptr, dinv, y2, zacc, NN, 128, 256);
  scatter_kernel<<<(NE * 32 + 255) / 256, 256, 0, stream>>>(
      y2, src, dst, zacc, 32, NE * 32);
  finish_kernel<<<(NN * 128 + 255) / 256, 256, 0, stream>>>(
      zacc, dinv, b2, 128, NN * 128);                        // z in zacc

  // --- projection head ---
  gemm_wmma_f32<2><<<dim3(NN / 16, 1), 256, 0, stream>>>(    // p = elu(z@fcW1+fcb1)
      zacc, fcW1, fcb1, nullptr, bufB, nullptr, NN, 128, 128);
  gemm_wmma_f32<1><<<dim3(NN / 16, 1), 256, 0, stream>>>(    // out = p@fcW2+fcb2
      bufB, fcW2, fcb2, nullptr, outp, nullptr, NN, 128, 128);
}